// Attention_36386962932368
// MI455X (gfx1250) — compile-verified
//
#include <hip/hip_runtime.h>
#include <hip/hip_bf16.h>

#ifndef __has_builtin
#define __has_builtin(x) 0
#endif

// FP32 WMMA building blocks for gfx1250 (wave32).
typedef float v2f __attribute__((ext_vector_type(2)));
typedef float v4f __attribute__((ext_vector_type(4)));
typedef float v8f __attribute__((ext_vector_type(8)));
typedef int   v4i __attribute__((ext_vector_type(4)));

#define WMMA_F32(a, b, c) \
    __builtin_amdgcn_wmma_f32_16x16x4_f32(false, (a), false, (b), (short)0, (c), false, false)
#define LO2(v) __builtin_shufflevector((v), (v), 0, 1)
#define HI2(v) __builtin_shufflevector((v), (v), 2, 3)

constexpr int Bn = 8;
constexpr int Ln = 2048;
constexpr int Dn = 768;
constexpr int Rn = Bn * Ln;       // 16384 rows of x
constexpr int DT = Dn / 16;       // 48

// k-permutation dual-WMMA step: one b128 per fragment covers 8 k-columns.
// WMMA#1 uses phys cols {k0,k0+1,k0+4,k0+5}, WMMA#2 {k0+2,k0+3,k0+6,k0+7};
// A and B share the same permutation, so the accumulated sum is exact.
#define STEP8(acc, a4, b4)                      \
    do {                                        \
        acc = WMMA_F32(LO2(a4), LO2(b4), acc);  \
        acc = WMMA_F32(HI2(a4), HI2(b4), acc);  \
    } while (0)

// --- CDNA5 async global->LDS copy (ASYNCcnt) with safe fallback -------------
#if __has_builtin(__builtin_amdgcn_global_load_async_to_lds_b128)
#define HAVE_ASYNC_LDS 1
typedef const __attribute__((address_space(1))) void* gas_void;
typedef __attribute__((address_space(3))) void*       las_void;
typedef __attribute__((address_space(1))) v4i*        gas_v4i;
typedef __attribute__((address_space(3))) v4i*        las_v4i;
__device__ __forceinline__ void lds_copy16(float* dst, const float* src) {
    __builtin_amdgcn_global_load_async_to_lds_b128(
        (gas_v4i)(gas_void)(const void*)src,
        (las_v4i)(las_void)(void*)dst,
        0, 0);
}
__device__ __forceinline__ void lds_copy_wait() {
#if __has_builtin(__builtin_amdgcn_s_wait_asynccnt)
    __builtin_amdgcn_s_wait_asynccnt(0);
#else
    asm volatile("s_wait_asynccnt 0" ::: "memory");
#endif
}
#else
#define HAVE_ASYNC_LDS 0
__device__ __forceinline__ void lds_copy16(float* dst, const float* src) {
    *(v4f*)dst = *(const v4f*)src;     // global_load_b128 + ds_store_b128
}
__device__ __forceinline__ void lds_copy_wait() {}  // barrier's dscnt wait covers it
#endif

// ---------------------------------------------------------------------------
// Kernel 1: fused QKV projection.
// WG = 8 waves covering 256 rows x 16 cols; all waves share one tn, so the
// three 16-row weight strips are staged in LDS (chunks of 128 k-columns,
// double-buffered, async-copy overlapped with WMMA) and fragments are read
// with ds_load_b128.  x fragments stay wave-private global b128 loads.
// Q stored row-major; K,V stored transposed per batch ([b][h][l]) so stage 2
// reads its reduction dim contiguously.
// ---------------------------------------------------------------------------
constexpr int KC  = 128;              // k-chunk
constexpr int LDW = KC + 4;           // padded LDS row stride (bank-conflict free)
constexpr int MATS = 16 * LDW;        // floats per staged matrix strip

__global__ __launch_bounds__(256) void qkv_wmma(
    const float* __restrict__ x,
    const float* __restrict__ Wq, const float* __restrict__ bq,
    const float* __restrict__ Wk, const float* __restrict__ bk,
    const float* __restrict__ Wv, const float* __restrict__ bv,
    float* __restrict__ Q, float* __restrict__ KT, float* __restrict__ VT)
{
    __shared__ float wlds[2][3 * MATS];           // 2 * 3 * 16 * 132 * 4B = 50.7 KB

    const int lane = threadIdx.x & 31;
    const int wave = threadIdx.x >> 5;
    const int tn = blockIdx.x % DT;               // 16-col block [0, 48)
    const int rg = blockIdx.x / DT;               // 256-row group [0, 64)
    const int tm = rg * 8 + wave;                 // this wave's 32-row block

    const int mr = lane & 15;
    const int h4 = (lane >> 4) << 2;              // {0, 4}

    const float* x0 = x + (size_t)(tm * 32 + mr) * Dn;
    const float* x1 = x0 + (size_t)16 * Dn;
    const float* wsrc[3] = { Wq + (size_t)tn * 16 * Dn,
                             Wk + (size_t)tn * 16 * Dn,
                             Wv + (size_t)tn * 16 * Dn };

    // Cooperative staging of one 3x16x128 chunk: 6 x b128 per thread.
    auto stage = [&](int c, int bufi) {
        const int t = threadIdx.x;
        #pragma unroll
        for (int j = 0; j < 2; ++j) {
            const int idx = t + 256 * j;          // 0..511 v4f slots per matrix
            const int row = idx >> 5;             // [0,16)
            const int cv  = (idx & 31) << 2;      // [0,128) step 4
            #pragma unroll
            for (int m = 0; m < 3; ++m)
                lds_copy16(&wlds[bufi][m * MATS + row * LDW + cv],
                           wsrc[m] + (size_t)row * Dn + c * KC + cv);
        }
    };

    v8f aq0 = {}, aq1 = {}, ak0 = {}, ak1 = {}, av0 = {}, av1 = {};
    int buf = 0;
    stage(0, 0);

    constexpr int NC = Dn / KC;                   // 6 chunks
    for (int c = 0; c < NC; ++c) {
        lds_copy_wait();                          // this wave's slice of chunk c landed
        __syncthreads();                          // whole chunk c visible; buf^1 free
        if (c + 1 < NC) {
            stage(c + 1, buf ^ 1);                // overlaps with WMMA below
            __builtin_prefetch(x0 + (c + 1) * KC, 0, 0);
            __builtin_prefetch(x1 + (c + 1) * KC, 0, 0);
        }
        const float* wp = &wlds[buf][mr * LDW];
        #pragma unroll 4
        for (int kk = 0; kk < KC; kk += 8) {
            const v4f a0 = *(const v4f*)(x0 + c * KC + kk + h4);
            const v4f a1 = *(const v4f*)(x1 + c * KC + kk + h4);
            const v4f fq = *(const v4f*)(wp + 0 * MATS + kk + h4);   // ds_load_b128
            const v4f fk = *(const v4f*)(wp + 1 * MATS + kk + h4);
            const v4f fv = *(const v4f*)(wp + 2 * MATS + kk + h4);
            STEP8(aq0, a0, fq);  STEP8(aq1, a1, fq);
            STEP8(ak0, a0, fk);  STEP8(ak1, a1, fk);
            STEP8(av0, a0, fv);  STEP8(av1, a1, fv);
        }
        buf ^= 1;
    }

    // C/D layout: lane = n + 16*(m>=8); VGPR r -> m = (lane>>4)*8 + r.
    const int n   = lane & 15;
    const int m0  = (lane >> 4) << 3;
    const int col = tn * 16 + n;
    const float biasq = bq[col], biask = bk[col], biasv = bv[col];

    const int b  = (tm * 32) / Ln;                // 32 | 2048: never straddles
    const int l0 = tm * 32 - b * Ln + m0;
    float* KTb = KT + (size_t)b * Dn * Ln + (size_t)col * Ln;
    float* VTb = VT + (size_t)b * Dn * Ln + (size_t)col * Ln;

    #pragma unroll
    for (int sub = 0; sub < 2; ++sub) {
        const v8f q = sub ? aq1 : aq0;
        const v8f k = sub ? ak1 : ak0;
        const v8f v = sub ? av1 : av0;
        #pragma unroll
        for (int r = 0; r < 8; ++r)
            Q[(size_t)(tm * 32 + sub * 16 + m0 + r) * Dn + col] = q[r] + biasq;
        const int lb = l0 + sub * 16;
        *(v4f*)(KTb + lb)     = (v4f){k[0] + biask, k[1] + biask, k[2] + biask, k[3] + biask};
        *(v4f*)(KTb + lb + 4) = (v4f){k[4] + biask, k[5] + biask, k[6] + biask, k[7] + biask};
        *(v4f*)(VTb + lb)     = (v4f){v[0] + biasv, v[1] + biasv, v[2] + biasv, v[3] + biasv};
        *(v4f*)(VTb + lb + 4) = (v4f){v[4] + biasv, v[5] + biasv, v[6] + biasv, v[7] + biasv};
    }
}

// ---------------------------------------------------------------------------
// Kernel 2: per-batch M = K^T V (768x768, reduction over l = 2048).
// Transposed operands -> both fragments contiguous along l: 4 b128 loads per
// 8 WMMAs, 32x32 tile per wave.  Output stored as M^T ([b][n][m]).
// ---------------------------------------------------------------------------
__global__ __launch_bounds__(256) void ktv_wmma(
    const float* __restrict__ KT, const float* __restrict__ VT,
    float* __restrict__ MT)
{
    const int lane = threadIdx.x & 31;
    const int wave = threadIdx.x >> 5;
    const int tile = blockIdx.x * 8 + wave;   // 8 * 24 * 24 tiles
    const int b  = tile / (24 * 24);
    const int t  = tile % (24 * 24);
    const int tm = t / 24;
    const int tn = t % 24;

    const int mr = lane & 15;
    const int h4 = (lane >> 4) << 2;

    const float* ka0 = KT + (size_t)b * Dn * Ln + (size_t)(tm * 32 + mr) * Ln;
    const float* ka1 = ka0 + (size_t)16 * Ln;
    const float* vb0 = VT + (size_t)b * Dn * Ln + (size_t)(tn * 32 + mr) * Ln;
    const float* vb1 = vb0 + (size_t)16 * Ln;

    v8f c00 = {}, c01 = {}, c10 = {}, c11 = {};
    #pragma unroll 4
    for (int kk = 0; kk < Ln; kk += 8) {
        const v4f a0 = *(const v4f*)(ka0 + kk + h4);
        const v4f a1 = *(const v4f*)(ka1 + kk + h4);
        const v4f f0 = *(const v4f*)(vb0 + kk + h4);
        const v4f f1 = *(const v4f*)(vb1 + kk + h4);
        STEP8(c00, a0, f0);  STEP8(c01, a0, f1);
        STEP8(c10, a1, f0);  STEP8(c11, a1, f1);
    }

    const int n  = lane & 15;
    const int m0 = (lane >> 4) << 3;
    float* MTb = MT + (size_t)b * Dn * Dn;
    #pragma unroll
    for (int ms = 0; ms < 2; ++ms) {
        #pragma unroll
        for (int ns = 0; ns < 2; ++ns) {
            const v8f c = ms ? (ns ? c11 : c10) : (ns ? c01 : c00);
            const int ng = tn * 32 + ns * 16 + n;
            const int mg = tm * 32 + ms * 16 + m0;
            *(v4f*)(MTb + (size_t)ng * Dn + mg)     = (v4f){c[0], c[1], c[2], c[3]};
            *(v4f*)(MTb + (size_t)ng * Dn + mg + 4) = (v4f){c[4], c[5], c[6], c[7]};
        }
    }
}

// ---------------------------------------------------------------------------
// Kernel 3: out = Q @ M * (1/sqrt(768)).  A = Q row-major, B from M^T; both
// contiguous along the reduction dim.  32x32 per wave, 4 b128 loads / 8 WMMAs.
// ---------------------------------------------------------------------------
__global__ __launch_bounds__(256) void qm_wmma(
    const float* __restrict__ Q, const float* __restrict__ MT,
    float* __restrict__ out)
{
    const int lane = threadIdx.x & 31;
    const int wave = threadIdx.x >> 5;
    const int tile = blockIdx.x * 8 + wave;   // 8 * 64 * 24 tiles
    const int b  = tile / (64 * 24);
    const int t  = tile % (64 * 24);
    const int tm = t / 24;
    const int tn = t % 24;

    const int mr = lane & 15;
    const int h4 = (lane >> 4) << 2;

    const float* qa0 = Q  + (size_t)b * Ln * Dn + (size_t)(tm * 32 + mr) * Dn;
    const float* qa1 = qa0 + (size_t)16 * Dn;
    const float* mb0 = MT + (size_t)b * Dn * Dn + (size_t)(tn * 32 + mr) * Dn;
    const float* mb1 = mb0 + (size_t)16 * Dn;

    v8f c00 = {}, c01 = {}, c10 = {}, c11 = {};
    #pragma unroll 4
    for (int kk = 0; kk < Dn; kk += 8) {
        const v4f a0 = *(const v4f*)(qa0 + kk + h4);
        const v4f a1 = *(const v4f*)(qa1 + kk + h4);
        const v4f f0 = *(const v4f*)(mb0 + kk + h4);
        const v4f f1 = *(const v4f*)(mb1 + kk + h4);
        STEP8(c00, a0, f0);  STEP8(c01, a0, f1);
        STEP8(c10, a1, f0);  STEP8(c11, a1, f1);
    }

    const float alpha = 0.03608439182435161f;   // 1/sqrt(768)
    const int n  = lane & 15;
    const int m0 = (lane >> 4) << 3;
    float* ob = out + (size_t)b * Ln * Dn;
    #pragma unroll
    for (int ms = 0; ms < 2; ++ms) {
        #pragma unroll
        for (int ns = 0; ns < 2; ++ns) {
            const v8f c = ms ? (ns ? c11 : c10) : (ns ? c01 : c00);
            const int ng = tn * 32 + ns * 16 + n;
            const int mg = tm * 32 + ms * 16 + m0;
            #pragma unroll
            for (int r = 0; r < 8; ++r)
                ob[(size_t)(mg + r) * Dn + ng] = c[r] * alpha;
        }
    }
}

// ---------------------------------------------------------------------------
extern "C" void kernel_launch(void* const* d_in, const int* in_sizes, int n_in,
                              void* d_out, int out_size, void* d_ws, size_t ws_size,
                              hipStream_t stream) {
    const float* x  = (const float*)d_in[0];
    const float* Wq = (const float*)d_in[1];
    const float* bq = (const float*)d_in[2];
    const float* Wk = (const float*)d_in[3];
    const float* bk = (const float*)d_in[4];
    const float* Wv = (const float*)d_in[5];
    const float* bv = (const float*)d_in[6];
    float* out = (float*)d_out;

    // Workspace: Q | K^T | V^T | M^T   (3*16384*768 + 8*768*768 floats = 169.9 MB)
    float* Q  = (float*)d_ws;
    float* KT = Q  + (size_t)Rn * Dn;
    float* VT = KT + (size_t)Rn * Dn;
    float* MT = VT + (size_t)Rn * Dn;

    dim3 blk(256);  // 8 waves per workgroup
    qkv_wmma<<<(Rn / 256) * DT, blk, 0, stream>>>(x, Wq, bq, Wk, bk, Wv, bv, Q, KT, VT);
    ktv_wmma<<<Bn * 24 * 24 / 8, blk, 0, stream>>>(KT, VT, MT);
    qm_wmma <<<Bn * 64 * 24 / 8, blk, 0, stream>>>(Q, MT, out);
}